// LuongAttnDecoderRNN_55757265436947
// MI455X (gfx1250) — compile-verified
//
#include <hip/hip_runtime.h>
#include <hip/hip_bf16.h>
#include <math.h>

typedef __attribute__((ext_vector_type(2))) float v2f;
typedef __attribute__((ext_vector_type(8))) float v8f;

#define V_SZ 50257
#define E_SZ 512
#define H_SZ 512
#define B_SZ 64
#define S_SZ 2048
#define G3H  (3 * H_SZ)

#define CHUNKS  16
#define S_CHUNK (S_SZ / CHUNKS)   // 128 s-values per partial block
#define TILE_S  8                 // s-tile staged in LDS (8 x 512 f32 = 16 KB)

// workspace offsets (in floats)
#define GI_OFF   0
#define GH_OFF   (GI_OFF + B_SZ * G3H)
#define HN_OFF   (GH_OFF + B_SZ * G3H)
#define EN_OFF   (HN_OFF + B_SZ * H_SZ)
#define CTX_OFF  (EN_OFF + B_SZ * S_SZ)
#define CC_OFF   (CTX_OFF + B_SZ * H_SZ)
#define PM_OFF   (CC_OFF + B_SZ * H_SZ)
#define PL_OFF   (PM_OFF + B_SZ * CHUNKS)
#define PCTX_OFF (PL_OFF + B_SZ * CHUNKS)
#define FM_OFF   (PCTX_OFF + B_SZ * CHUNKS * H_SZ)
#define FL_OFF   (FM_OFF + B_SZ)

// output offsets (in floats): output [B,V], hidden [1,B,H], attn [B,1,S]
#define OUT_OUT  0
#define OUT_HID  (B_SZ * V_SZ)
#define OUT_ATTN (OUT_HID + B_SZ * H_SZ)

__device__ __forceinline__ v8f wmma_f32x4(v2f a, v2f b, v8f c) {
    // D = A(16x4,f32) * B(4x16,f32) + C(16x16,f32)
    return __builtin_amdgcn_wmma_f32_16x16x4_f32(
        /*neg_a=*/false, a, /*neg_b=*/false, b,
        /*c_mod=*/(short)0, c, /*reuse_a=*/false, /*reuse_b=*/false);
}

// ---------------------------------------------------------------------------
// Kernel 1: GRU input GEMMs, gi = x @ w_ih^T + b_ih ; gh = h @ w_hh^T + b_hh
// One wave (32 threads) per 16x16 output tile; embedding lookup fused into A.
// ---------------------------------------------------------------------------
__global__ __launch_bounds__(32)
void gru_gemm_kernel(const int* __restrict__ seq, const float* __restrict__ emb,
                     const float* __restrict__ h_prev,
                     const float* __restrict__ w_ih, const float* __restrict__ w_hh,
                     const float* __restrict__ b_ih, const float* __restrict__ b_hh,
                     float* __restrict__ ws) {
    const int lane = threadIdx.x;
    const int hi   = lane >> 4;
    const int lr   = lane & 15;
    const int n0   = blockIdx.x * 16;  // gate column tile (0..1535)
    const int m0   = blockIdx.y * 16;  // batch row tile (0..63)
    const int m    = m0 + lr;
    const int n    = n0 + lr;

    const long ridx = (long)seq[m];
    const float* xrow = emb + ridx * (long)E_SZ;
    const float* hrow = h_prev + (long)m * H_SZ;
    const float* w1   = w_ih + (long)n * E_SZ;
    const float* w2   = w_hh + (long)n * H_SZ;

    v8f acc1, acc2;
    const float bi = b_ih[n], bh = b_hh[n];
    for (int v = 0; v < 8; ++v) { acc1[v] = bi; acc2[v] = bh; }

    for (int k = 0; k < H_SZ; k += 4) {
        const int kk = k + 2 * hi;
        v2f a1, a2, bb1, bb2;
        a1.x = xrow[kk];  a1.y = xrow[kk + 1];
        a2.x = hrow[kk];  a2.y = hrow[kk + 1];
        bb1.x = w1[kk];   bb1.y = w1[kk + 1];
        bb2.x = w2[kk];   bb2.y = w2[kk + 1];
        acc1 = wmma_f32x4(a1, bb1, acc1);
        acc2 = wmma_f32x4(a2, bb2, acc2);
    }

    float* gi = ws + GI_OFF;
    float* gh = ws + GH_OFF;
    for (int v = 0; v < 8; ++v) {
        const int row = m0 + v + 8 * hi;
        gi[row * G3H + n] = acc1[v];
        gh[row * G3H + n] = acc2[v];
    }
}

// ---------------------------------------------------------------------------
// Kernel 2: GRU gates (PyTorch order [r, z, n]) -> h_new, also hidden output
// ---------------------------------------------------------------------------
__global__ __launch_bounds__(256)
void gates_kernel(const float* __restrict__ h_prev, const float* __restrict__ ws_gi,
                  const float* __restrict__ ws_gh, float* __restrict__ h_new,
                  float* __restrict__ out_hidden) {
    const int j = blockIdx.x * blockDim.x + threadIdx.x;
    if (j >= B_SZ * H_SZ) return;
    const int b  = j / H_SZ;
    const int hc = j % H_SZ;
    const float* gi = ws_gi + b * G3H;
    const float* gh = ws_gh + b * G3H;
    const float i_r = gi[hc], i_z = gi[H_SZ + hc], i_n = gi[2 * H_SZ + hc];
    const float h_r = gh[hc], h_z = gh[H_SZ + hc], h_n = gh[2 * H_SZ + hc];
    const float r  = 1.0f / (1.0f + expf(-(i_r + h_r)));
    const float z  = 1.0f / (1.0f + expf(-(i_z + h_z)));
    const float nn = tanhf(i_n + r * h_n);
    const float hv = (1.0f - z) * nn + z * h_prev[j];
    h_new[j] = hv;
    out_hidden[j] = hv;
}

// ---------------------------------------------------------------------------
// Kernel 3: fused flash-style attention partial.
// Block (b, chunk): streams its 128 s-values of enc ONCE through a 16 KB LDS
// tile; computes energies, online-softmax running (m,l), and the weighted
// context accumulator from the same tile. Writes raw energies + partials.
// ---------------------------------------------------------------------------
__global__ __launch_bounds__(256)
void attn_partial_kernel(const float* __restrict__ h_new, const float* __restrict__ enc,
                         float* __restrict__ energies, float* __restrict__ part_m,
                         float* __restrict__ part_l, float* __restrict__ part_ctx) {
    const int c   = blockIdx.x;        // chunk index (0..CHUNKS-1)
    const int b   = blockIdx.y;        // batch row
    const int s0  = c * S_CHUNK;
    const int tid = threadIdx.x;
    const int wv  = tid >> 5;          // wave id (0..7)
    const int lane = tid & 31;

    __shared__ float hb_s[H_SZ];               // 2 KB: query row
    __shared__ float tile[TILE_S * H_SZ];      // 16 KB: enc s-tile
    __shared__ float es[TILE_S];               // tile energies

    for (int i = tid; i < H_SZ; i += 256) hb_s[i] = h_new[b * H_SZ + i];
    __syncthreads();

    float m_run = -3.402823466e+38f;
    float l_run = 0.0f;
    float ctx0 = 0.0f, ctx1 = 0.0f;            // h = tid, h = tid + 256

    for (int t = 0; t < S_CHUNK; t += TILE_S) {
        // Coalesced float4 load of 8 x 512 enc values into LDS.
        for (int i = 0; i < (TILE_S * H_SZ / 4) / 256; ++i) {
            const int idx = tid + 256 * i;             // 0..1023
            const int row = idx >> 7;                  // /128 float4 per row
            const int kq  = idx & 127;
            const long g = (long)(s0 + t + row) * B_SZ * H_SZ + (long)b * H_SZ + 4 * kq;
            const float4 v = *reinterpret_cast<const float4*>(enc + g);
            *reinterpret_cast<float4*>(tile + row * H_SZ + 4 * kq) = v;
        }
        __syncthreads();

        // Each of the 8 waves computes the dot for one s of the tile.
        float p = 0.0f;
        const float* tr = tile + wv * H_SZ;
        for (int h = lane; h < H_SZ; h += 32) p = fmaf(hb_s[h], tr[h], p);
        for (int off = 16; off > 0; off >>= 1) p += __shfl_down(p, off, 32);
        if (lane == 0) {
            es[wv] = p;
            energies[b * S_SZ + s0 + t + wv] = p;
        }
        __syncthreads();

        // Online softmax update (computed redundantly by all threads).
        float m_new = m_run;
        for (int j = 0; j < TILE_S; ++j) m_new = fmaxf(m_new, es[j]);
        const float scale = expf(m_run - m_new);
        float wj[TILE_S];
        float lsum = 0.0f;
        for (int j = 0; j < TILE_S; ++j) { wj[j] = expf(es[j] - m_new); lsum += wj[j]; }
        l_run = l_run * scale + lsum;
        ctx0 *= scale;
        ctx1 *= scale;
        for (int j = 0; j < TILE_S; ++j) {
            ctx0 = fmaf(wj[j], tile[j * H_SZ + tid], ctx0);
            ctx1 = fmaf(wj[j], tile[j * H_SZ + tid + 256], ctx1);
        }
        m_run = m_new;
        __syncthreads();   // tile reused next iteration
    }

    const int pc = b * CHUNKS + c;
    if (tid == 0) { part_m[pc] = m_run; part_l[pc] = l_run; }
    part_ctx[(long)pc * H_SZ + tid]       = ctx0;
    part_ctx[(long)pc * H_SZ + tid + 256] = ctx1;
}

// ---------------------------------------------------------------------------
// Kernel 4: combine per-chunk partials into final context and global (M, L).
// ---------------------------------------------------------------------------
__global__ __launch_bounds__(256)
void attn_combine_kernel(const float* __restrict__ part_m, const float* __restrict__ part_l,
                         const float* __restrict__ part_ctx, float* __restrict__ ctx,
                         float* __restrict__ fm, float* __restrict__ fl) {
    const int b   = blockIdx.x;
    const int tid = threadIdx.x;

    float M = -3.402823466e+38f;
    for (int i = 0; i < CHUNKS; ++i) M = fmaxf(M, part_m[b * CHUNKS + i]);
    float L = 0.0f;
    float c0 = 0.0f, c1 = 0.0f;
    for (int i = 0; i < CHUNKS; ++i) {
        const int pc = b * CHUNKS + i;
        const float w = expf(part_m[pc] - M);
        L = fmaf(w, part_l[pc], L);
        c0 = fmaf(w, part_ctx[(long)pc * H_SZ + tid], c0);
        c1 = fmaf(w, part_ctx[(long)pc * H_SZ + tid + 256], c1);
    }
    const float invL = 1.0f / L;
    ctx[b * H_SZ + tid]       = c0 * invL;
    ctx[b * H_SZ + tid + 256] = c1 * invL;
    if (tid == 0) { fm[b] = M; fl[b] = L; }
}

// ---------------------------------------------------------------------------
// Kernel 5: attention weights output: attn = exp(e - M) / L
// ---------------------------------------------------------------------------
__global__ __launch_bounds__(256)
void attn_weights_kernel(const float* __restrict__ energies, const float* __restrict__ fm,
                         const float* __restrict__ fl, float* __restrict__ attn) {
    const int j = blockIdx.x * blockDim.x + threadIdx.x;
    if (j >= B_SZ * S_SZ) return;
    const int b = j / S_SZ;
    attn[j] = expf(energies[j] - fm[b]) / fl[b];
}

// ---------------------------------------------------------------------------
// Kernel 6: concat_output = tanh([h_new, ctx] @ w_concat^T + b_concat)  (WMMA)
// ---------------------------------------------------------------------------
__global__ __launch_bounds__(32)
void concat_gemm_kernel(const float* __restrict__ h_new, const float* __restrict__ ctxv,
                        const float* __restrict__ w_concat, const float* __restrict__ b_concat,
                        float* __restrict__ cc) {
    const int lane = threadIdx.x;
    const int hi   = lane >> 4;
    const int lr   = lane & 15;
    const int n0   = blockIdx.x * 16;
    const int m0   = blockIdx.y * 16;
    const int m    = m0 + lr;
    const int n    = n0 + lr;

    const float* a1row = h_new + (long)m * H_SZ;
    const float* a2row = ctxv  + (long)m * H_SZ;
    const float* wrow  = w_concat + (long)n * (2 * H_SZ);

    v8f acc;
    const float bc = b_concat[n];
    for (int v = 0; v < 8; ++v) acc[v] = bc;

    for (int k = 0; k < H_SZ; k += 4) {
        const int kk = k + 2 * hi;
        v2f a, bb;
        a.x  = a1row[kk];  a.y  = a1row[kk + 1];
        bb.x = wrow[kk];   bb.y = wrow[kk + 1];
        acc = wmma_f32x4(a, bb, acc);
    }
    for (int k = 0; k < H_SZ; k += 4) {
        const int kk = k + 2 * hi;
        v2f a, bb;
        a.x  = a2row[kk];        a.y  = a2row[kk + 1];
        bb.x = wrow[H_SZ + kk];  bb.y = wrow[H_SZ + kk + 1];
        acc = wmma_f32x4(a, bb, acc);
    }

    for (int v = 0; v < 8; ++v) {
        const int row = m0 + v + 8 * hi;
        cc[row * H_SZ + n] = tanhf(acc[v]);
    }
}

// ---------------------------------------------------------------------------
// Kernel 7: output = concat_output @ w_out^T + b_out   (WMMA, 16x64 per wave)
// ---------------------------------------------------------------------------
__global__ __launch_bounds__(32)
void out_gemm_kernel(const float* __restrict__ cc, const float* __restrict__ w_out,
                     const float* __restrict__ b_out, float* __restrict__ out) {
    const int lane = threadIdx.x;
    const int hi   = lane >> 4;
    const int lr   = lane & 15;
    const int n0   = blockIdx.x * 64;
    const int m0   = blockIdx.y * 16;

    const float* arow = cc + (long)(m0 + lr) * H_SZ;

    v8f acc[4];
    const float* wrow[4];
    float sc[4];
    int nidx[4];
    for (int j = 0; j < 4; ++j) {
        const int n = n0 + 16 * j + lr;
        nidx[j] = n;
        const bool ok = (n < V_SZ);
        sc[j] = ok ? 1.0f : 0.0f;
        wrow[j] = w_out + (long)(ok ? n : 0) * H_SZ;
        const float bv = ok ? b_out[n] : 0.0f;
        for (int v = 0; v < 8; ++v) acc[j][v] = bv;
    }

    for (int k = 0; k < H_SZ; k += 4) {
        const int kk = k + 2 * hi;
        v2f a;
        a.x = arow[kk];
        a.y = arow[kk + 1];
        for (int j = 0; j < 4; ++j) {
            v2f bb;
            bb.x = wrow[j][kk] * sc[j];
            bb.y = wrow[j][kk + 1] * sc[j];
            acc[j] = wmma_f32x4(a, bb, acc[j]);
        }
    }

    for (int j = 0; j < 4; ++j) {
        const int n = nidx[j];
        if (n < V_SZ) {
            for (int v = 0; v < 8; ++v) {
                const int m = m0 + v + 8 * hi;
                out[(long)m * V_SZ + n] = acc[j][v];
            }
        }
    }
}

// ---------------------------------------------------------------------------
extern "C" void kernel_launch(void* const* d_in, const int* in_sizes, int n_in,
                              void* d_out, int out_size, void* d_ws, size_t ws_size,
                              hipStream_t stream) {
    const int*   seq         = (const int*)  d_in[0];
    const float* last_hidden = (const float*)d_in[1];
    const float* enc         = (const float*)d_in[2];
    const float* emb         = (const float*)d_in[3];
    const float* w_ih        = (const float*)d_in[4];
    const float* w_hh        = (const float*)d_in[5];
    const float* b_ih        = (const float*)d_in[6];
    const float* b_hh        = (const float*)d_in[7];
    const float* w_concat    = (const float*)d_in[8];
    const float* b_concat    = (const float*)d_in[9];
    const float* w_out       = (const float*)d_in[10];
    const float* b_out       = (const float*)d_in[11];
    float* out = (float*)d_out;
    float* ws  = (float*)d_ws;

    // 1) GRU GEMMs (WMMA)
    gru_gemm_kernel<<<dim3(G3H / 16, B_SZ / 16), 32, 0, stream>>>(
        seq, emb, last_hidden, w_ih, w_hh, b_ih, b_hh, ws);
    // 2) gates -> h_new (+ hidden output)
    gates_kernel<<<(B_SZ * H_SZ + 255) / 256, 256, 0, stream>>>(
        last_hidden, ws + GI_OFF, ws + GH_OFF, ws + HN_OFF, out + OUT_HID);
    // 3) fused attention partials: enc streamed exactly once (256 MB total)
    attn_partial_kernel<<<dim3(CHUNKS, B_SZ), 256, 0, stream>>>(
        ws + HN_OFF, enc, ws + EN_OFF, ws + PM_OFF, ws + PL_OFF, ws + PCTX_OFF);
    // 4) combine partials -> context, global (M, L)
    attn_combine_kernel<<<B_SZ, 256, 0, stream>>>(
        ws + PM_OFF, ws + PL_OFF, ws + PCTX_OFF, ws + CTX_OFF, ws + FM_OFF, ws + FL_OFF);
    // 5) attention weights output
    attn_weights_kernel<<<(B_SZ * S_SZ + 255) / 256, 256, 0, stream>>>(
        ws + EN_OFF, ws + FM_OFF, ws + FL_OFF, out + OUT_ATTN);
    // 6) concat GEMM + tanh (WMMA)
    concat_gemm_kernel<<<dim3(H_SZ / 16, B_SZ / 16), 32, 0, stream>>>(
        ws + HN_OFF, ws + CTX_OFF, w_concat, b_concat, ws + CC_OFF);
    // 7) vocab projection (WMMA)
    out_gemm_kernel<<<dim3((V_SZ + 63) / 64, B_SZ / 16), 32, 0, stream>>>(
        ws + CC_OFF, w_out, b_out, out + OUT_OUT);
}